// SCC_26491358282237
// MI455X (gfx1250) — compile-verified
//
#include <hip/hip_runtime.h>

typedef float v2f __attribute__((ext_vector_type(2)));
typedef float v8f __attribute__((ext_vector_type(8)));

#define C_IN   64
#define C_OUT  64
#define Gwin   8
#define STEP   4
#define PIX    16384                 // H*W = 128*128
#define NBATCH 32
#define STRIPS_PER_BATCH 1024        // PIX/16
#define TOTAL_STRIPS (NBATCH * STRIPS_PER_BATCH)   // 32768

// D = A(16x4) * B(4x16) + C, fp32 exact, one 16-pixel strip per wave,
// full 64-channel output per strip via 4 M-tiles x 16 K-steps of WMMA.
__global__ __launch_bounds__(256) void scc_wmma_f32_kernel(
    const float* __restrict__ x,      // (32, 64, 128, 128)
    const float* __restrict__ w,      // (64, 8)
    float* __restrict__ out)          // (32, 64, 128, 128)
{
    __shared__ float wfull[C_OUT * C_IN];   // 16 KB dense scattered weights

    const int tid = threadIdx.x;

    // Build W_full in LDS: zero then scatter (indices within a row are
    // distinct 8 consecutive mod-64 channels, so plain stores suffice).
    for (int i = tid; i < C_OUT * C_IN; i += 256) wfull[i] = 0.0f;
    __syncthreads();
    if (tid < C_OUT * Gwin) {
        const int o = tid >> 3;
        const int j = tid & 7;
        const int c = (o * STEP + j) & (C_IN - 1);
        wfull[o * C_IN + c] = w[tid];
    }
    __syncthreads();

    const int lane   = tid & 31;
    const int wave   = tid >> 5;
    const int laneLo = lane & 15;          // N index / M index within half
    const int laneHi = lane >> 4;          // 0: lanes 0-15, 1: lanes 16-31
    const int totalWaves = gridDim.x * 8;

    for (int s = blockIdx.x * 8 + wave; s < TOTAL_STRIPS; s += totalWaves) {
        const int b  = s >> 10;            // batch
        const int p0 = (s & 1023) << 4;    // pixel base within batch

        const float* xb = x   + b * (C_IN  * PIX) + p0 + laneLo;
        float*       ob = out + b * (C_OUT * PIX) + p0 + laneLo;

        // ---- Load all B tiles for this strip: x element touched once ----
        // B (4x16, KxN): VGPR0 = K=kb+0 (lanes 0-15) / K=kb+2 (lanes 16-31),
        //                VGPR1 = K=kb+1 / K=kb+3. N = laneLo.
        v2f Bt[16];
        #pragma unroll
        for (int k16 = 0; k16 < 16; ++k16) {
            const int kc = k16 * 4 + 2 * laneHi;     // channel for this half
            Bt[k16].x = __builtin_nontemporal_load(&xb[(kc + 0) * PIX]);
            Bt[k16].y = __builtin_nontemporal_load(&xb[(kc + 1) * PIX]);
        }

        // ---- 4 M-tiles of 16 output channels ----
        #pragma unroll
        for (int m = 0; m < 4; ++m) {
            const int o0 = m * 16;
            v8f acc = {};
            // A (16x4): lane<16 -> M=laneLo, K=k,k+1 ; lane>=16 -> K=k+2,k+3
            const float* wrow = &wfull[(o0 + laneLo) * C_IN + 2 * laneHi];
            #pragma unroll
            for (int k16 = 0; k16 < 16; ++k16) {
                v2f a;
                a.x = wrow[k16 * 4 + 0];   // merges to ds_load_b64 (8B aligned)
                a.y = wrow[k16 * 4 + 1];
                acc = __builtin_amdgcn_wmma_f32_16x16x4_f32(
                          /*neg_a=*/false, a,
                          /*neg_b=*/false, Bt[k16],
                          /*c_mod=*/(short)0, acc,
                          /*reuse_a=*/false, /*reuse_b=*/false);
            }
            // C/D (16x16): VGPR r -> row o0+r (lanes 0-15) / o0+r+8 (16-31)
            #pragma unroll
            for (int r = 0; r < 8; ++r) {
                __builtin_nontemporal_store(acc[r],
                    &ob[(o0 + r + 8 * laneHi) * PIX]);
            }
        }
    }
}

extern "C" void kernel_launch(void* const* d_in, const int* in_sizes, int n_in,
                              void* d_out, int out_size, void* d_ws, size_t ws_size,
                              hipStream_t stream) {
    const float* x = (const float*)d_in[0];   // (32,64,128,128) fp32
    const float* w = (const float*)d_in[1];   // (64,8) fp32
    float* out     = (float*)d_out;           // (32,64,128,128) fp32
    (void)in_sizes; (void)n_in; (void)out_size; (void)d_ws; (void)ws_size;

    dim3 grid(1024);   // 8 waves/block -> 8192 waves, 4 strips each
    dim3 block(256);
    scc_wmma_f32_kernel<<<grid, block, 0, stream>>>(x, w, out);
}